// RadiusGraph_47416438948014
// MI455X (gfx1250) — compile-verified
//
#include <hip/hip_runtime.h>
#include <hip/hip_bf16.h>

typedef __attribute__((ext_vector_type(2))) float v2f;
typedef __attribute__((ext_vector_type(8))) float v8f;

#define BS   4
#define P0   16384
#define P1   4096
#define KNB  32
#define R2   0.0625f   // 0.25^2

// ---------------------------------------------------------------------------
// Wave32 argmax reduction (value, index), ties -> lowest index (matches argmax)
// ---------------------------------------------------------------------------
__device__ __forceinline__ void wave_argmax(float& v, int& i) {
#pragma unroll
    for (int off = 16; off >= 1; off >>= 1) {
        float ov = __shfl_xor(v, off, 32);
        int   oi = __shfl_xor(i, off, 32);
        if (ov > v || (ov == v && oi < i)) { v = ov; i = oi; }
    }
}

// ---------------------------------------------------------------------------
// Kernel 1: furthest point sampling. One workgroup (1024 thr = 32 waves) per
// batch. Point cloud lives in LDS SoA (192 KB, needs CDNA5's 320 KB WGP LDS),
// min_d lives in registers (16 per thread).
// ---------------------------------------------------------------------------
__global__ void fps_kernel(const float* __restrict__ xyz,
                           int* __restrict__ ctr_idx,
                           float* __restrict__ child_xyz) {
    extern __shared__ float smem[];
    float* px   = smem;            // P0
    float* py   = px + P0;         // P0
    float* pz   = py + P0;         // P0
    float* redv = pz + P0;         // 32
    int*   redi = (int*)(redv + 32);   // 32
    float* bc   = (float*)(redi + 32); // 4 (x,y,z,spare)

    const int b = blockIdx.x;
    const int t = threadIdx.x;
    const float* xb = xyz + (size_t)b * P0 * 3;

    for (int i = t; i < P0; i += 1024) {
        px[i] = xb[i * 3 + 0];
        py[i] = xb[i * 3 + 1];
        pz[i] = xb[i * 3 + 2];
    }
    __syncthreads();

    float mind[16];
#pragma unroll
    for (int j = 0; j < 16; ++j) mind[j] = 1e10f;

    // sample 0 is point 0 (reference starts at index 0)
    float sx = px[0], sy = py[0], sz = pz[0];
    if (t == 0) {
        ctr_idx[b * P1 + 0] = 0;
        child_xyz[((size_t)b * P1 + 0) * 3 + 0] = sx;
        child_xyz[((size_t)b * P1 + 0) * 3 + 1] = sy;
        child_xyz[((size_t)b * P1 + 0) * 3 + 2] = sz;
    }

    const int lane = t & 31;
    const int wave = t >> 5;

    for (int s = 1; s < P1; ++s) {
        float bv = -1.0f;
        int   bi = 0;
#pragma unroll
        for (int j = 0; j < 16; ++j) {
            int   idx = t + j * 1024;
            float dx = px[idx] - sx, dy = py[idx] - sy, dz = pz[idx] - sz;
            float d  = dx * dx + dy * dy + dz * dz;
            float m  = mind[j] < d ? mind[j] : d;
            mind[j]  = m;
            if (m > bv) { bv = m; bi = idx; }   // strict > : first (lowest) idx wins
        }
        wave_argmax(bv, bi);
        if (lane == 0) { redv[wave] = bv; redi[wave] = bi; }
        __syncthreads();
        if (wave == 0) {
            float v  = redv[lane];
            int   i2 = redi[lane];
            wave_argmax(v, i2);
            if (lane == 0) {
                float cx = px[i2], cy = py[i2], cz = pz[i2];
                bc[0] = cx; bc[1] = cy; bc[2] = cz;
                ctr_idx[b * P1 + s] = i2;
                child_xyz[((size_t)b * P1 + s) * 3 + 0] = cx;
                child_xyz[((size_t)b * P1 + s) * 3 + 1] = cy;
                child_xyz[((size_t)b * P1 + s) * 3 + 2] = cz;
            }
        }
        __syncthreads();
        sx = bc[0]; sy = bc[1]; sz = bc[2];
    }
}

// ---------------------------------------------------------------------------
// Kernel 2: ball query via WMMA f32 16x16x4.
// Block = 256 thr (8 waves); each wave owns 16 centers. Points (SoA + |p|^2,
// 256 KB) staged in LDS once per block. Per 16-point tile: one
// v_wmma_f32_16x16x4_f32 gives the 16x16 dot tile; wave32 ballots over d2<r^2
// give per-row 16-bit hit masks. Ballot words are wave-uniform, so an
// all-zero tile is skipped with a single SALU test (common case ~94% of
// row/tile pairs are empty). Lanes 0..15 peel bits in ascending index order.
// ---------------------------------------------------------------------------
__global__ void ballquery_kernel(const float* __restrict__ xyz,
                                 const int* __restrict__ ctr_idx,
                                 float* __restrict__ out_edges,
                                 float* __restrict__ out_filled) {
    extern __shared__ float smem[];
    float* pxs = smem;
    float* pys = pxs + P0;
    float* pzs = pys + P0;
    float* p2s = pzs + P0;

    const int b = blockIdx.y;
    const int t = threadIdx.x;
    const float* xb = xyz + (size_t)b * P0 * 3;

    for (int i = t; i < P0; i += 256) {
        float x = xb[i * 3 + 0], y = xb[i * 3 + 1], z = xb[i * 3 + 2];
        pxs[i] = x; pys[i] = y; pzs[i] = z;
        p2s[i] = x * x + y * y + z * z;
    }
    __syncthreads();

    const int  lane  = t & 31;
    const int  wave  = t >> 5;
    const int  cbase = blockIdx.x * 128 + wave * 16;   // 16 centers per wave
    const int  m     = lane & 15;
    const bool hi    = lane >= 16;

    const int* cib = ctr_idx + b * P1 + cbase;

    // branchless lane-half split: hi lanes read z from the pzs region
    const float* base0 = hi ? pzs : pxs;   // per-lane base (cndmask on address)
    const float  ymul  = hi ? 0.0f : 1.0f; // kills the y element on hi lanes

    // A fragment (16x4, K-pad with 0): lanes 0-15 carry {x,y}, 16-31 carry {z,0}
    const int myCtrPoint = cib[m];
    v2f A;
    A.x = base0[myCtrPoint];
    A.y = pys[myCtrPoint] * ymul;

    // |c|^2 for the 8 D rows this lane sees (rows v / v+8)
    float c2v[8];
#pragma unroll
    for (int v = 0; v < 8; ++v) {
        int row = v + (hi ? 8 : 0);
        c2v[v] = p2s[cib[row]];
    }

    int   cnt      = hi ? KNB : 0;   // lanes >=16 never extract; pre-satisfied
    int   firstIdx = 0;
    float* oe = out_edges  + ((size_t)(b * P1 + cbase) * KNB) * 2;
    float* of = out_filled +  (size_t)(b * P1 + cbase) * KNB;

    for (int n0 = 0; n0 < P0; n0 += 16) {
        const int n = n0 + m;
        v2f Bf;
        Bf.x = base0[n];
        Bf.y = pys[n] * ymul;
        const float x2n = p2s[n];

        v8f C = {};
        v8f D = __builtin_amdgcn_wmma_f32_16x16x4_f32(
            false, A, false, Bf, (short)0, C, false, false);

        unsigned w[8];
#pragma unroll
        for (int v = 0; v < 8; ++v) {
            float d2 = c2v[v] + x2n - 2.0f * D[v];
            w[v] = __builtin_amdgcn_ballot_w32(d2 < R2);
        }

        // wave-uniform (SALU) skip of tiles with no hits for any of the 16 rows
        const unsigned any = w[0] | w[1] | w[2] | w[3] | w[4] | w[5] | w[6] | w[7];
        if (any) {
            if (lane < 16) {
                unsigned full = w[lane & 7];
                unsigned mask = (lane < 8) ? (full & 0xFFFFu) : (full >> 16);
                while (mask && cnt < KNB) {
                    int bit = __builtin_ctz(mask);
                    mask &= mask - 1;
                    int pidx = n0 + bit;
                    if (cnt == 0) firstIdx = pidx;
                    int slot = cnt++;
                    oe[(lane * KNB + slot) * 2 + 0] = (float)pidx;
                    oe[(lane * KNB + slot) * 2 + 1] = (float)myCtrPoint;
                    of[lane * KNB + slot]           = 1.0f;
                }
            }
            // uniform early exit once every row has K neighbors (cnt only
            // changes on non-empty tiles, so the check lives here)
            unsigned done = __builtin_amdgcn_ballot_w32(cnt >= KNB);
            if (done == 0xFFFFFFFFu) break;
        }
    }

    // unfilled slots replicate first hit (or 0), is_filled = false
    if (lane < 16) {
        for (int slot = cnt; slot < KNB; ++slot) {
            oe[(lane * KNB + slot) * 2 + 0] = (float)firstIdx;
            oe[(lane * KNB + slot) * 2 + 1] = (float)myCtrPoint;
            of[lane * KNB + slot]           = 0.0f;
        }
    }
}

// ---------------------------------------------------------------------------
extern "C" void kernel_launch(void* const* d_in, const int* in_sizes, int n_in,
                              void* d_out, int out_size, void* d_ws, size_t ws_size,
                              hipStream_t stream) {
    const float* xyz = (const float*)d_in[0];

    float* out        = (float*)d_out;
    float* out_edges  = out;                                   // (bs,p1,K,2)
    float* out_filled = out + (size_t)BS * P1 * KNB * 2;       // (bs,p1,K)
    float* out_child  = out_filled + (size_t)BS * P1 * KNB;    // (bs,p1,3)

    int* ctr_ws = (int*)d_ws;                                  // (bs,p1) int32

    const size_t fpsLds = (size_t)3 * P0 * sizeof(float)
                        + 32 * sizeof(float) + 32 * sizeof(int) + 4 * sizeof(float);
    fps_kernel<<<BS, 1024, fpsLds, stream>>>(xyz, ctr_ws, out_child);

    const size_t bqLds = (size_t)4 * P0 * sizeof(float);
    ballquery_kernel<<<dim3(P1 / 128, BS), 256, bqLds, stream>>>(
        xyz, ctr_ws, out_edges, out_filled);
}